// SimpleCompPCFGCharNoDistinction_23184233464139
// MI455X (gfx1250) — compile-verified
//
#include <hip/hip_runtime.h>
#include <math.h>

// ---------------- problem constants (from reference) ----------------
// P=4, D=3  ->  trees(3) = 4 + 2*36*36 = 2596, C = 2597
static constexpr int Pn   = 4;
static constexpr int Bn   = 4;
static constexpr int Nn   = 20;
static constexpr int Vn   = 10000;
static constexpr int SDn  = 64;
static constexpr int Cn   = 2597;
static constexpr int CPAD = 2608;          // 163 * 16, WMMA row tiles
static constexpr float NEGV = -1000000000.0f;

typedef float v2f __attribute__((ext_vector_type(2)));
typedef float v8f __attribute__((ext_vector_type(8)));

__device__ __forceinline__ float lae(float a, float b) {
    float mx = fmaxf(a, b);
    float mn = fminf(a, b);
    return mx + log1pf(expf(mn - mx));
}

// ---------------------------------------------------------------------------
// K0: per nonterminal i, scan column i of rule weight + filter matrices.
//     lw[i] = logsoftmax_row_i(scores)[c1(i)]   (split0 added later in K1)
//     leftL[i] = c1(i) or -1.  Same for right rules.
// Column scans are coalesced across the wave (address j*C + i).
// ---------------------------------------------------------------------------
__global__ __launch_bounds__(256) void k_grammar_scan(
    const float* __restrict__ Wl, const float* __restrict__ bl, const float* __restrict__ fl,
    const float* __restrict__ Wr, const float* __restrict__ br, const float* __restrict__ fr,
    float* __restrict__ lw, float* __restrict__ rw,
    int* __restrict__ leftL, int* __restrict__ leftR)
{
    int i = blockIdx.x * blockDim.x + threadIdx.x;
    if (i >= Cn) return;

    // left rules
    {
        float m = -3.0e38f, s = 0.0f, sc = 0.0f; int c1 = -1;
        for (int j = 0; j < Cn; ++j) {
            float v = Wl[(size_t)j * Cn + i] + bl[j];
            if (v > m) { s = s * expf(m - v) + 1.0f; m = v; }
            else       { s += expf(v - m); }
            if (fl[(size_t)j * Cn + i] > -1.0e6f) { c1 = j; sc = v; }
        }
        float lse = m + logf(s);
        leftL[i] = c1;
        lw[i] = (c1 >= 0) ? (sc - lse) : 0.0f;
    }
    // right rules
    {
        float m = -3.0e38f, s = 0.0f, sc = 0.0f; int c1 = -1;
        for (int j = 0; j < Cn; ++j) {
            float v = Wr[(size_t)j * Cn + i] + br[j];
            if (v > m) { s = s * expf(m - v) + 1.0f; m = v; }
            else       { s += expf(v - m); }
            if (fr[(size_t)j * Cn + i] > -1.0e6f) { c1 = j; sc = v; }
        }
        float lse = m + logf(s);
        leftR[i] = c1;
        rw[i] = (c1 >= 0) ? (sc - lse) : 0.0f;
    }
}

// ---------------------------------------------------------------------------
// K1: split MLP + relu_he + root pre-scores; pads relu_he rows [C, CPAD) with 0.
// ---------------------------------------------------------------------------
__global__ __launch_bounds__(256) void k_mlp(
    const float* __restrict__ sW1, const float* __restrict__ sb1,
    const float* __restrict__ r1Wa, const float* __restrict__ r1ba,
    const float* __restrict__ r1Wb, const float* __restrict__ r1bb,
    const float* __restrict__ r2Wa, const float* __restrict__ r2ba,
    const float* __restrict__ r2Wb, const float* __restrict__ r2bb,
    const float* __restrict__ Wo,   const float* __restrict__ bo,
    const float* __restrict__ eW1,  const float* __restrict__ eb1,
    const float* __restrict__ rootW, const float* __restrict__ rootb,
    const float* __restrict__ rootf,
    float* __restrict__ relu_he, float* __restrict__ split0, float* __restrict__ split1,
    float* __restrict__ lw, float* __restrict__ rw, float* __restrict__ rs_pre)
{
    int i = blockIdx.x * blockDim.x + threadIdx.x;
    if (i >= CPAD) return;
    if (i >= Cn) {                       // zero pad rows for WMMA tiles
        for (int k = 0; k < SDn; ++k) relu_he[(size_t)i * SDn + k] = 0.0f;
        return;
    }

    float h[SDn], t[SDn], u[SDn];
    for (int k = 0; k < SDn; ++k) h[k] = sW1[(size_t)k * Cn + i] + sb1[k];

    // residual block 1
    for (int k = 0; k < SDn; ++k) {
        float a = r1ba[k];
        for (int j = 0; j < SDn; ++j) a += r1Wa[k * SDn + j] * h[j];
        t[k] = fmaxf(a, 0.0f);
    }
    for (int k = 0; k < SDn; ++k) {
        float a = r1bb[k];
        for (int j = 0; j < SDn; ++j) a += r1Wb[k * SDn + j] * t[j];
        u[k] = fmaxf(a, 0.0f);
    }
    for (int k = 0; k < SDn; ++k) h[k] += u[k];
    // residual block 2
    for (int k = 0; k < SDn; ++k) {
        float a = r2ba[k];
        for (int j = 0; j < SDn; ++j) a += r2Wa[k * SDn + j] * h[j];
        t[k] = fmaxf(a, 0.0f);
    }
    for (int k = 0; k < SDn; ++k) {
        float a = r2bb[k];
        for (int j = 0; j < SDn; ++j) a += r2Wb[k * SDn + j] * t[j];
        u[k] = fmaxf(a, 0.0f);
    }
    for (int k = 0; k < SDn; ++k) h[k] += u[k];

    float o0 = bo[0], o1 = bo[1];
    for (int j = 0; j < SDn; ++j) { o0 += Wo[j] * h[j]; o1 += Wo[SDn + j] * h[j]; }
    float mo = fmaxf(o0, o1);
    float ls = mo + logf(expf(o0 - mo) + expf(o1 - mo));
    float s0 = o0 - ls, s1 = o1 - ls;
    split0[i] = s0; split1[i] = s1;
    lw[i] += s0;                        // finalize exp-matrix log-entries
    rw[i] += s0;

    for (int k = 0; k < SDn; ++k)
        relu_he[(size_t)i * SDn + k] = fmaxf(eW1[(size_t)k * Cn + i] + eb1[k], 0.0f);

    rs_pre[i] = rootf[i] + rootW[i] + rootb[i];   // root_emb is 1x1 identity
}

// ---------------------------------------------------------------------------
// K3: emission row-logsumexp via V_WMMA_F32_16X16X4_F32.
// One wave per 16-row tile of relu_he; sweeps V in 16-column tiles,
// GEMM K=64 done as 16 WMMA steps of K=4; online logsumexp per row.
// A (16x4 f32): VGPR0/1 hold K=0/1 on lanes 0-15, K=2/3 on lanes 16-31.
// B (4x16 f32): VGPR0 = rows K=0 (lanes 0-15) / K=1 (lanes 16-31),
//               VGPR1 = rows K=2 / K=3.   C/D: VGPRr = row r / r+8.
// ---------------------------------------------------------------------------
__global__ __launch_bounds__(32) void k_emit_lse(
    const float* __restrict__ he,   // [CPAD][64]
    const float* __restrict__ W2,   // [V][64]
    const float* __restrict__ b2,   // [V]
    float* __restrict__ row_lse)    // [CPAD]
{
    const int lane = threadIdx.x;
    const int c0   = blockIdx.x * 16;
    const int half = (lane < 16) ? 0 : 1;
    const int lrow = lane & 15;

    // preload this wave's A-fragment K-pairs for all 16 K-steps
    float2 afrag[16];
    {
        const float* ap = he + (size_t)(c0 + lrow) * SDn + 2 * half;
        #pragma unroll
        for (int kk = 0; kk < 16; ++kk)
            afrag[kk] = *(const float2*)(ap + 4 * kk);
    }

    float mx[8], sm[8];
    #pragma unroll
    for (int r = 0; r < 8; ++r) { mx[r] = -3.0e38f; sm[r] = 0.0f; }

    for (int v0 = 0; v0 < Vn; v0 += 16) {
        const float* wp = W2 + (size_t)(v0 + lrow) * SDn;
        v8f acc = {};
        #pragma unroll
        for (int kk = 0; kk < 16; ++kk) {
            float4 w4 = *(const float4*)(wp + 4 * kk);
            v2f a; a[0] = afrag[kk].x; a[1] = afrag[kk].y;
            v2f b; b[0] = half ? w4.y : w4.x;   // K = 4kk + (0|1)
                   b[1] = half ? w4.w : w4.z;   // K = 4kk + (2|3)
            acc = __builtin_amdgcn_wmma_f32_16x16x4_f32(
                false, a, false, b, (short)0, acc, false, false);
        }
        float bias = b2[v0 + lrow];
        #pragma unroll
        for (int r = 0; r < 8; ++r) {
            float x = acc[r] + bias;            // s[row, v0+lrow]
            if (x > mx[r]) { sm[r] = sm[r] * expf(mx[r] - x) + 1.0f; mx[r] = x; }
            else           { sm[r] += expf(x - mx[r]); }
        }
    }

    // merge the 16 column-lanes of each half-wave
    #pragma unroll
    for (int msk = 1; msk < 16; msk <<= 1) {
        #pragma unroll
        for (int r = 0; r < 8; ++r) {
            float om = __shfl_xor(mx[r], msk, 32);
            float os = __shfl_xor(sm[r], msk, 32);
            float nm = fmaxf(mx[r], om);
            sm[r] = sm[r] * expf(mx[r] - nm) + os * expf(om - nm);
            mx[r] = nm;
        }
    }
    if (lane == 0) {
        #pragma unroll
        for (int r = 0; r < 8; ++r) row_lse[c0 + r] = mx[r] + logf(sm[r]);
    } else if (lane == 16) {
        #pragma unroll
        for (int r = 0; r < 8; ++r) row_lse[c0 + 8 + r] = mx[r] + logf(sm[r]);
    }
}

// ---------------------------------------------------------------------------
// K4: leaf scores -> beta[w=1]:  beta[0][n][b][c] = logdist[c, x[b,n]] + split1[c]
// ---------------------------------------------------------------------------
__global__ __launch_bounds__(256) void k_leaf(
    const int* __restrict__ x, const float* __restrict__ he,
    const float* __restrict__ W2, const float* __restrict__ b2,
    const float* __restrict__ row_lse, const float* __restrict__ split1,
    float* __restrict__ beta)
{
    __shared__ float wrow[SDn];
    const int tok = blockIdx.x;            // 0 .. B*N-1
    const int b = tok / Nn, n = tok % Nn;
    const int xv = x[b * Nn + n];
    if (threadIdx.x < SDn) wrow[threadIdx.x] = W2[(size_t)xv * SDn + threadIdx.x];
    __syncthreads();
    const float bias = b2[xv];
    for (int i = threadIdx.x; i < Cn; i += blockDim.x) {
        float d = bias;
        const float* hp = he + (size_t)i * SDn;
        for (int k = 0; k < SDn; ++k) d += hp[k] * wrow[k];
        beta[((size_t)(0 * Nn + n) * Bn + b) * Cn + i] = d - row_lse[i] + split1[i];
    }
}

// ---------------------------------------------------------------------------
// K5: one inside step (span width w). Exact-in-f32 sparse form of the
// reference's dense logmm: row max m taken over the full dense u vector,
// then column i gets its single surviving term + 1e-38 floor.
// Left/right beta rows are staged into LDS via the CDNA5 async DMA path
// (GLOBAL_LOAD_ASYNC_TO_LDS_B32, tracked by ASYNCcnt) when available; the
// next split point's rows are prefetched (global_prefetch_b8) meanwhile.
// B32 is used because C=2597 makes row bases only 4-byte aligned.
// ---------------------------------------------------------------------------
static constexpr int NACC = (Cn + 255) / 256;

#if defined(__gfx1250__) && __has_builtin(__builtin_amdgcn_global_load_async_to_lds_b32)
#define USE_ASYNC_LDS 1
#else
#define USE_ASYNC_LDS 0
#endif

typedef __attribute__((address_space(1))) int* gas_i32p;
typedef __attribute__((address_space(3))) int* las_i32p;

__global__ __launch_bounds__(256) void k_inside(
    float* __restrict__ beta,
    const int* __restrict__ l2r, const int* __restrict__ r2l,
    const int* __restrict__ leftL, const int* __restrict__ leftR,
    const float* __restrict__ lw, const float* __restrict__ rw,
    int w)
{
    __shared__ __align__(16) float Ls[Cn];
    __shared__ __align__(16) float Rs[Cn];
    __shared__ int   l2rs[Cn];
    __shared__ int   r2ls[Cn];
    __shared__ float red1[256];
    __shared__ float red2[256];

    const int tid = threadIdx.x;
    const int s = blockIdx.x / Bn;
    const int b = blockIdx.x % Bn;

    for (int j = tid; j < Cn; j += 256) { l2rs[j] = l2r[j]; r2ls[j] = r2l[j]; }

    float acc[NACC];
    #pragma unroll
    for (int t = 0; t < NACC; ++t) acc[t] = NEGV;

    for (int k = 1; k < w; ++k) {
        const float* Lp = beta + ((size_t)((k - 1) * Nn + s) * Bn + b) * Cn;
        const float* Rp = beta + ((size_t)((w - k - 1) * Nn + (s + k)) * Bn + b) * Cn;
        __syncthreads();                               // protect LDS reuse
#if USE_ASYNC_LDS
        for (int j = tid; j < Cn; j += 256) {
            __builtin_amdgcn_global_load_async_to_lds_b32(
                (gas_i32p)(Lp + j), (las_i32p)&Ls[j], 0, 0);
            __builtin_amdgcn_global_load_async_to_lds_b32(
                (gas_i32p)(Rp + j), (las_i32p)&Rs[j], 0, 0);
        }
#else
        for (int j = tid; j < Cn; j += 256) { Ls[j] = Lp[j]; Rs[j] = Rp[j]; }
#endif
        // prefetch next split point's rows while the async DMA runs
        if (k + 1 < w && tid < 82) {
            const float* Lnx = beta + ((size_t)(k * Nn + s) * Bn + b) * Cn;
            const float* Rnx = beta + ((size_t)((w - k - 2) * Nn + (s + k + 1)) * Bn + b) * Cn;
            __builtin_prefetch(Lnx + tid * 32, 0, 3);
            __builtin_prefetch(Rnx + tid * 32, 0, 3);
        }
#if USE_ASYNC_LDS
  #if __has_builtin(__builtin_amdgcn_s_wait_asynccnt)
        __builtin_amdgcn_s_wait_asynccnt(0);
  #else
        asm volatile("s_wait_asynccnt 0" ::: "memory");
  #endif
#endif
        __syncthreads();

        float pm1 = -3.0e38f, pm2 = -3.0e38f;
        for (int j = tid; j < Cn; j += 256) {
            pm1 = fmaxf(pm1, Ls[j] + Rs[l2rs[j]]);
            pm2 = fmaxf(pm2, Ls[r2ls[j]] + Rs[j]);
        }
        red1[tid] = pm1; red2[tid] = pm2;
        __syncthreads();
        for (int off = 128; off > 0; off >>= 1) {
            if (tid < off) {
                red1[tid] = fmaxf(red1[tid], red1[tid + off]);
                red2[tid] = fmaxf(red2[tid], red2[tid + off]);
            }
            __syncthreads();
        }
        const float m1 = red1[0], m2 = red2[0];

        int t = 0;
        for (int i = tid; i < Cn; i += 256, ++t) {
            int jl = leftL[i];
            int jr = leftR[i];
            float e1 = (jl >= 0) ? expf(Ls[jl] + Rs[l2rs[jl]] - m1) * expf(lw[i]) : 0.0f;
            float e2 = (jr >= 0) ? expf(Ls[r2ls[jr]] + Rs[jr] - m2) * expf(rw[i]) : 0.0f;
            float v1 = logf(e1 + 1e-38f) + m1;
            float v2 = logf(e2 + 1e-38f) + m2;
            acc[t] = lae(acc[t], lae(v1, v2));
        }
    }

    __syncthreads();
    int t = 0;
    for (int i = tid; i < Cn; i += 256, ++t)
        beta[((size_t)((w - 1) * Nn + s) * Bn + b) * Cn + i] = acc[t];
}

// ---------------------------------------------------------------------------
// K6: out[b] = -( lse_i(beta_top + rs_pre) - lse_i(rs_pre) )
// ---------------------------------------------------------------------------
__global__ __launch_bounds__(256) void k_final(
    const float* __restrict__ beta, const float* __restrict__ rs, float* __restrict__ out)
{
    __shared__ float red[256];
    const int b = blockIdx.x, tid = threadIdx.x;
    const float* top = beta + ((size_t)((Nn - 1) * Nn + 0) * Bn + b) * Cn;

    // lse1 over top+rs
    float pm = -3.0e38f;
    for (int i = tid; i < Cn; i += 256) pm = fmaxf(pm, top[i] + rs[i]);
    red[tid] = pm; __syncthreads();
    for (int off = 128; off > 0; off >>= 1) { if (tid < off) red[tid] = fmaxf(red[tid], red[tid + off]); __syncthreads(); }
    const float m1 = red[0]; __syncthreads();
    float ps = 0.0f;
    for (int i = tid; i < Cn; i += 256) ps += expf(top[i] + rs[i] - m1);
    red[tid] = ps; __syncthreads();
    for (int off = 128; off > 0; off >>= 1) { if (tid < off) red[tid] += red[tid + off]; __syncthreads(); }
    const float s1 = red[0]; __syncthreads();

    // lse2 over rs (root log-softmax normalizer)
    pm = -3.0e38f;
    for (int i = tid; i < Cn; i += 256) pm = fmaxf(pm, rs[i]);
    red[tid] = pm; __syncthreads();
    for (int off = 128; off > 0; off >>= 1) { if (tid < off) red[tid] = fmaxf(red[tid], red[tid + off]); __syncthreads(); }
    const float m2 = red[0]; __syncthreads();
    ps = 0.0f;
    for (int i = tid; i < Cn; i += 256) ps += expf(rs[i] - m2);
    red[tid] = ps; __syncthreads();
    for (int off = 128; off > 0; off >>= 1) { if (tid < off) red[tid] += red[tid + off]; __syncthreads(); }
    const float s2 = red[0];

    if (tid == 0) out[b] = -((m1 + logf(s1)) - (m2 + logf(s2)));
}

// ---------------------------------------------------------------------------
extern "C" void kernel_launch(void* const* d_in, const int* in_sizes, int n_in,
                              void* d_out, int out_size, void* d_ws, size_t ws_size,
                              hipStream_t stream) {
    (void)in_sizes; (void)n_in; (void)out_size; (void)ws_size;

    const int*   x      = (const int*)  d_in[0];
    const float* rootW  = (const float*)d_in[3];
    const float* rootb  = (const float*)d_in[4];
    const float* ruleLW = (const float*)d_in[5];
    const float* ruleLb = (const float*)d_in[6];
    const float* ruleRW = (const float*)d_in[7];
    const float* ruleRb = (const float*)d_in[8];
    const float* sW1    = (const float*)d_in[9];
    const float* sb1    = (const float*)d_in[10];
    const float* r1Wa   = (const float*)d_in[11];
    const float* r1ba   = (const float*)d_in[12];
    const float* r1Wb   = (const float*)d_in[13];
    const float* r1bb   = (const float*)d_in[14];
    const float* r2Wa   = (const float*)d_in[15];
    const float* r2ba   = (const float*)d_in[16];
    const float* r2Wb   = (const float*)d_in[17];
    const float* r2bb   = (const float*)d_in[18];
    const float* Wo     = (const float*)d_in[19];
    const float* bo     = (const float*)d_in[20];
    const float* eW1    = (const float*)d_in[21];
    const float* eb1    = (const float*)d_in[22];
    const float* eW2    = (const float*)d_in[23];
    const float* eb2    = (const float*)d_in[24];
    const float* fl     = (const float*)d_in[25];
    const float* fr     = (const float*)d_in[26];
    const float* rootf  = (const float*)d_in[27];
    const int*   l2r    = (const int*)  d_in[28];
    const int*   r2l    = (const int*)  d_in[29];

    // workspace layout
    float* ws      = (float*)d_ws;
    float* relu_he = ws;                         // CPAD * 64
    float* row_lse = relu_he + (size_t)CPAD * SDn;  // CPAD
    float* split0  = row_lse + CPAD;             // C
    float* split1  = split0 + Cn;                // C
    float* lw      = split1 + Cn;                // C
    float* rw      = lw + Cn;                    // C
    float* rs_pre  = rw + Cn;                    // C
    int*   leftL   = (int*)(rs_pre + Cn);        // C
    int*   leftR   = leftL + Cn;                 // C
    float* beta    = (float*)(leftR + Cn);       // N*N*B*C

    const int tpb = 256;

    k_grammar_scan<<<(Cn + tpb - 1) / tpb, tpb, 0, stream>>>(
        ruleLW, ruleLb, fl, ruleRW, ruleRb, fr, lw, rw, leftL, leftR);

    k_mlp<<<(CPAD + tpb - 1) / tpb, tpb, 0, stream>>>(
        sW1, sb1, r1Wa, r1ba, r1Wb, r1bb, r2Wa, r2ba, r2Wb, r2bb,
        Wo, bo, eW1, eb1, rootW, rootb, rootf,
        relu_he, split0, split1, lw, rw, rs_pre);

    k_emit_lse<<<CPAD / 16, 32, 0, stream>>>(relu_he, eW2, eb2, row_lse);

    k_leaf<<<Bn * Nn, tpb, 0, stream>>>(x, relu_he, eW2, eb2, row_lse, split1, beta);

    for (int w = 2; w <= Nn; ++w) {
        const int S = Nn - w + 1;
        k_inside<<<S * Bn, tpb, 0, stream>>>(beta, l2r, r2l, leftL, leftR, lw, rw, w);
    }

    k_final<<<Bn, tpb, 0, stream>>>(beta, rs_pre, (float*)d_out);
}